// CropAndResize_3882650436632
// MI455X (gfx1250) — compile-verified
//
#include <hip/hip_runtime.h>
#include <hip/hip_bf16.h>
#include <stdint.h>
#include <stddef.h>

namespace {

constexpr int Cch  = 256;   // channels
constexpr int Him  = 160;   // image H
constexpr int Wim  = 256;   // image W
constexpr int Nbox = 256;   // number of boxes
constexpr int CH   = 32;    // crop H
constexpr int CW   = 32;    // crop W

// 16-byte interpolation table entry (one per crop row / crop column)
struct alignas(16) Entry {
    int   i0;   // low index (clamped)
    int   i1;   // high index (clamped)
    float fr;   // fractional weight
    float va;   // 1.0 if in-bounds else 0.0
};

__device__ __forceinline__ Entry make_entry(const float* __restrict__ boxes, int n, int t) {
    const float y1 = boxes[4 * n + 0], x1 = boxes[4 * n + 1];
    const float y2 = boxes[4 * n + 2], x2 = boxes[4 * n + 3];
    const bool  isX = (t < 32);
    const int   i   = isX ? t : (t - 32);
    const float c1  = isX ? x1 : y1;
    const float c2  = isX ? x2 : y2;
    const float Dm1 = isX ? (float)(Wim - 1) : (float)(Him - 1);
    const float scale = (c2 - c1) * Dm1 / (float)(CW - 1);
    const float pos   = c1 * Dm1 + (float)i * scale;
    Entry e;
    e.va = (pos >= 0.0f && pos <= Dm1) ? 1.0f : 0.0f;
    const float f0 = floorf(pos);
    e.fr = pos - f0;
    e.i0 = (int)fminf(fmaxf(f0,        0.0f), Dm1);
    e.i1 = (int)fminf(fmaxf(f0 + 1.0f, 0.0f), Dm1);
    return e;
}

// Kernel 1: build per-box tables in scratch. Layout per box n:
//   ws[n*64 + 0..31]  : X entries (l, r, lx, valid_x)
//   ws[n*64 + 32..63] : Y entries (t, b, ly, valid_y)
__global__ void precompute_tables(const float* __restrict__ boxes, Entry* __restrict__ ws) {
    ws[(size_t)blockIdx.x * 64 + threadIdx.x] = make_entry(boxes, blockIdx.x, threadIdx.x);
}

// Kernel 2: one block per (n, c); 256 threads; lane = crop-x, wave = base crop-y.
template <bool USE_ASYNC_WS>
__global__ void crop_resize_kernel(const float* __restrict__ image,
                                   const float* __restrict__ boxes,
                                   const int*   __restrict__ box_ind,
                                   const Entry* __restrict__ ws,
                                   float*       __restrict__ out) {
    __shared__ alignas(16) Entry sTab[64];   // [0..31]=X, [32..63]=Y

    const int c   = blockIdx.x;
    const int n   = blockIdx.y;
    const int tid = threadIdx.x;

    if (USE_ASYNC_WS) {
        // Stage this box's 1 KB table global->LDS with the CDNA5 async LDS DMA.
        // Wave 0: each lane moves 2x16B; offset:512 applies to BOTH the LDS and
        // global address (CDNA5 ISA), so one VGPR pair serves both ops.
        if (tid < 32) {
            const unsigned lds   = (unsigned)(size_t)(&sTab[tid]);     // low 32 bits of flat ptr = LDS byte addr
            const unsigned voff  = (unsigned)tid * (unsigned)sizeof(Entry);
            const unsigned long long gbase =
                (unsigned long long)(size_t)(ws + (size_t)n * 64);
            asm volatile(
                "global_load_async_to_lds_b128 %0, %1, %2\n\t"
                "global_load_async_to_lds_b128 %0, %1, %2 offset:512\n\t"
                "s_wait_asynccnt 0"
                :
                : "v"(lds), "v"(voff), "s"(gbase)
                : "memory");
        }
    } else {
        if (tid < 64) sTab[tid] = make_entry(boxes, n, tid);
    }
    __syncthreads();

    const int x  = tid & 31;   // lane -> crop x  (coalesced 128B stores per wave)
    const int y0 = tid >> 5;   // wave -> base crop y

    const int bi = box_ind[n];
    const float* __restrict__ plane = image + ((size_t)bi * Cch + c) * (size_t)(Him * Wim);
    float*       __restrict__ po    = out   + (((size_t)n * Cch + c) * CH) * CW;

    const Entry ex = sTab[x];

#pragma unroll
    for (int k = 0; k < 4; ++k) {
        const int   y  = y0 + (k << 3);
        const Entry ey = sTab[32 + y];

        if (k < 3) {
            // CDNA5 global_prefetch_b8 for next iteration's top row
            const Entry eyn = sTab[32 + y0 + ((k + 1) << 3)];
            __builtin_prefetch(plane + (size_t)eyn.i0 * Wim + ex.i0, 0, 0);
        }

        const float* rt = plane + (size_t)ey.i0 * Wim;
        const float* rb = plane + (size_t)ey.i1 * Wim;
        const float tl = rt[ex.i0];
        const float tr = rt[ex.i1];
        const float bl = rb[ex.i0];
        const float br = rb[ex.i1];

        const float top = fmaf(tr - tl, ex.fr, tl);
        const float bot = fmaf(br - bl, ex.fr, bl);
        float v         = fmaf(bot - top, ey.fr, top);
        v = (ex.va * ey.va != 0.0f) ? v : 0.0f;

        // Non-temporal store: don't let the 256MB output stream evict the
        // 160MB image from the 192MB L2 (keeps all gathers L2-resident).
        __builtin_nontemporal_store(v, po + (size_t)y * CW + x);
    }
}

} // anonymous namespace

extern "C" void kernel_launch(void* const* d_in, const int* in_sizes, int n_in,
                              void* d_out, int out_size, void* d_ws, size_t ws_size,
                              hipStream_t stream) {
    (void)in_sizes; (void)n_in; (void)out_size;

    const float* image   = (const float*)d_in[0];
    const float* boxes   = (const float*)d_in[1];
    const int*   box_ind = (const int*)d_in[2];
    float*       out     = (float*)d_out;

    const size_t ws_needed = (size_t)Nbox * 64 * sizeof(Entry);   // 256 KB
    const bool   use_ws    = (d_ws != nullptr) && (ws_size >= ws_needed);

    const dim3 grid(Cch, Nbox);
    const dim3 block(256);

    if (use_ws) {
        Entry* tab = (Entry*)d_ws;
        precompute_tables<<<Nbox, 64, 0, stream>>>(boxes, tab);
        crop_resize_kernel<true><<<grid, block, 0, stream>>>(image, boxes, box_ind, tab, out);
    } else {
        crop_resize_kernel<false><<<grid, block, 0, stream>>>(image, boxes, box_ind, nullptr, out);
    }
}